// NodeEncoder_73830487818322
// MI455X (gfx1250) — compile-verified
//
#include <hip/hip_runtime.h>
#include <hip/hip_bf16.h>

#define EMBED  768
#define LATENT 256

typedef __attribute__((ext_vector_type(2))) float v2f;
typedef __attribute__((ext_vector_type(8))) float v8f;

// ---------------------------------------------------------------------------
// GEMM: OUT[n, l] = sum_k X[n,k] * W[l,k] + bias[l]
// X: [Nrows, 768] row-major, W: [256, 768] row-major, OUT: [Nrows, 256].
// Block = 256 threads (8 waves). Block tile 128(M) x 64(N).
// Wave grid 4(M) x 2(N); each wave computes 32x32 via 2x2 WMMA f32 16x16x4.
// ---------------------------------------------------------------------------
__global__ __launch_bounds__(256)
void leconv_gemm_xwt_bias(const float* __restrict__ X,
                          const float* __restrict__ W,
                          const float* __restrict__ bias,
                          float* __restrict__ OUT,
                          int Nrows) {
    __shared__ float Xs[128 * 17];   // [row][k], stride 17 -> conflict-free
    __shared__ float Ws[64 * 17];

    const int tid  = threadIdx.x;
    const int wave = tid >> 5;       // 0..7
    const int lane = tid & 31;
    const int wm   = wave >> 1;      // 0..3 (M)
    const int wn   = wave & 1;       // 0..1 (N)
    const int l15  = lane & 15;
    const int hi   = lane >> 4;      // 0 or 1 (half-wave)

    const int block_m0 = blockIdx.x * 128;
    const int block_n0 = blockIdx.y * 64;

    v8f acc[2][2] = {};              // [mi][ni], zero-init

    const int r64 = tid >> 2;        // 0..63
    const int c4  = (tid & 3) * 4;   // 0,4,8,12

    for (int kc = 0; kc < EMBED; kc += 16) {
        // ---- stage X tile 128x16 (two float4 per thread, coalesced) ----
        #pragma unroll
        for (int it = 0; it < 2; ++it) {
            const int row  = r64 + it * 64;
            const int grow = block_m0 + row;
            float4 v = make_float4(0.f, 0.f, 0.f, 0.f);
            if (grow < Nrows)
                v = *(const float4*)(X + (size_t)grow * EMBED + kc + c4);
            float* d = &Xs[row * 17 + c4];
            d[0] = v.x; d[1] = v.y; d[2] = v.z; d[3] = v.w;
        }
        // ---- stage W tile 64x16 (one float4 per thread) ----
        {
            const int wl = block_n0 + r64;
            float4 v = *(const float4*)(W + (size_t)wl * EMBED + kc + c4);
            float* d = &Ws[r64 * 17 + c4];
            d[0] = v.x; d[1] = v.y; d[2] = v.z; d[3] = v.w;
        }
        __syncthreads();

        // ---- 4 WMMA K-steps of 4 over the 16-wide chunk ----
        #pragma unroll
        for (int k4 = 0; k4 < 16; k4 += 4) {
            const int ka = k4 + hi * 2;  // lanes 0-15: K={k4,k4+1}; 16-31: {k4+2,k4+3}
            v2f afr[2], bfr[2];
            #pragma unroll
            for (int mi = 0; mi < 2; ++mi) {
                const float* p = &Xs[(wm * 32 + mi * 16 + l15) * 17 + ka];
                afr[mi][0] = p[0]; afr[mi][1] = p[1];
            }
            #pragma unroll
            for (int ni = 0; ni < 2; ++ni) {
                const float* p = &Ws[(wn * 32 + ni * 16 + l15) * 17 + ka];
                bfr[ni][0] = p[0]; bfr[ni][1] = p[1];
            }
            #pragma unroll
            for (int mi = 0; mi < 2; ++mi)
                #pragma unroll
                for (int ni = 0; ni < 2; ++ni)
                    acc[mi][ni] = __builtin_amdgcn_wmma_f32_16x16x4_f32(
                        false, afr[mi], false, bfr[ni],
                        (short)0, acc[mi][ni], false, false);
        }
        __syncthreads();
    }

    // ---- store + bias. D layout: VGPR r -> M=r (lanes 0-15), M=r+8 (16-31) ----
    #pragma unroll
    for (int mi = 0; mi < 2; ++mi) {
        #pragma unroll
        for (int ni = 0; ni < 2; ++ni) {
            const int n  = block_n0 + wn * 32 + ni * 16 + l15;
            const float bv = bias ? bias[n] : 0.f;
            #pragma unroll
            for (int r = 0; r < 8; ++r) {
                const int m = block_m0 + wm * 32 + mi * 16 + r + hi * 8;
                if (m < Nrows)
                    OUT[(size_t)m * LATENT + n] = acc[mi][ni][r] + bv;
            }
        }
    }
}

// ---------------------------------------------------------------------------
__global__ __launch_bounds__(256)
void leconv_zero_f32(float* __restrict__ p, int n) {
    int i = blockIdx.x * 256 + threadIdx.x;
    if (i < n) p[i] = 0.f;
}

// deg[i] = sum of E_w over edges whose dst == i
__global__ __launch_bounds__(256)
void leconv_deg(const int* __restrict__ dst, const float* __restrict__ ew,
                float* __restrict__ deg, int E) {
    int e = blockIdx.x * 256 + threadIdx.x;
    if (e < E) atomicAdd(&deg[dst[e]], ew[e]);
}

// agg[dst] += ew * a[src]   (one wave per edge; lane covers 8 dims via 2x float4)
__global__ __launch_bounds__(256)
void leconv_edge(const int* __restrict__ src, const int* __restrict__ dst,
                 const float* __restrict__ ew, const float* __restrict__ a,
                 float* __restrict__ agg, int E) {
    const int e = blockIdx.x * 8 + (threadIdx.x >> 5);
    if (e >= E) return;
    const int lane = threadIdx.x & 31;
    const int s = src[e];
    const int d = dst[e];
    const float w = ew[e];
    const float4* ap = (const float4*)(a + (size_t)s * LATENT);
    float*        gp = agg + (size_t)d * LATENT;
    #pragma unroll
    for (int it = 0; it < 2; ++it) {
        const int i4 = lane + it * 32;        // 64 float4 per row
        const float4 v = ap[i4];
        atomicAdd(&gp[i4 * 4 + 0], w * v.x);
        atomicAdd(&gp[i4 * 4 + 1], w * v.y);
        atomicAdd(&gp[i4 * 4 + 2], w * v.z);
        atomicAdd(&gp[i4 * 4 + 3], w * v.w);
    }
}

// out = relu(agg - deg[n] * b)   (agg already holds c + edge sums)
__global__ __launch_bounds__(256)
void leconv_finalize(const float* __restrict__ agg, const float* __restrict__ bmat,
                     const float* __restrict__ deg, float* __restrict__ out,
                     int total) {
    int i = blockIdx.x * 256 + threadIdx.x;
    if (i < total) {
        const int n = i >> 8;                 // / LATENT
        const float v = agg[i] - deg[n] * bmat[i];
        out[i] = v > 0.f ? v : 0.f;
    }
}

// ---------------------------------------------------------------------------
extern "C" void kernel_launch(void* const* d_in, const int* in_sizes, int n_in,
                              void* d_out, int out_size, void* d_ws, size_t ws_size,
                              hipStream_t stream) {
    const float* x    = (const float*)d_in[0];
    const int*   Eidx = (const int*)  d_in[1];   // [2, E] flat
    const float* Ew   = (const float*)d_in[2];
    const float* W1   = (const float*)d_in[3];
    const float* b1   = (const float*)d_in[4];
    const float* W2   = (const float*)d_in[5];
    const float* W3   = (const float*)d_in[6];
    const float* b3   = (const float*)d_in[7];

    const int N = in_sizes[0] / EMBED;           // 50000
    const int E = in_sizes[2];                   // 800000
    const int* src = Eidx;
    const int* dst = Eidx + E;

    float* a_buf = (float*)d_ws;                 // [N, 256]  lin1
    float* b_buf = a_buf + (size_t)N * LATENT;   // [N, 256]  lin2
    float* agg   = b_buf + (size_t)N * LATENT;   // [N, 256]  lin3 + edge sums
    float* deg   = agg   + (size_t)N * LATENT;   // [N]
    float* out   = (float*)d_out;

    dim3 gblk(256);
    dim3 ggrid((N + 127) / 128, LATENT / 64);

    // a = x@W1^T + b1 ; b = x@W2^T ; agg = c = x@W3^T + b3
    leconv_gemm_xwt_bias<<<ggrid, gblk, 0, stream>>>(x, W1, b1,      a_buf, N);
    leconv_gemm_xwt_bias<<<ggrid, gblk, 0, stream>>>(x, W2, nullptr, b_buf, N);
    leconv_gemm_xwt_bias<<<ggrid, gblk, 0, stream>>>(x, W3, b3,      agg,   N);

    leconv_zero_f32<<<(N + 255) / 256, 256, 0, stream>>>(deg, N);
    leconv_deg<<<(E + 255) / 256, 256, 0, stream>>>(dst, Ew, deg, E);

    leconv_edge<<<(E + 7) / 8, 256, 0, stream>>>(src, dst, Ew, a_buf, agg, E);

    const int total = N * LATENT;
    leconv_finalize<<<(total + 255) / 256, 256, 0, stream>>>(agg, b_buf, deg, out, total);
}